// FaNat_35854386987345
// MI455X (gfx1250) — compile-verified
//
#include <hip/hip_runtime.h>
#include <stdint.h>

// CTC forced alignment (Viterbi) + backtrace + trigger mask for MI455X (gfx1250).
//
// One workgroup per batch element. The time loop is a serial scan; the scattered
// vocab gather (257 x f32 per step) is hidden with CDNA5 async global->LDS copies
// (GLOBAL_LOAD_ASYNC_TO_LDS_B32, ASYNCcnt) pipelined 3 steps deep across 4 LDS
// buffers. Backpointers (2 bits/state/step) for the whole utterance are packed
// into LDS (~64KB) -- feasible because CDNA5 WGPs have 320KB LDS -- so the serial
// backtrace never touches HBM.

#define LOGZERO (-1.0e10f)
#define LMAX 272          // >= L = 2*ymax+1 = 257, padded
#define TMAX 1024         // xmax
#define TB4  (TMAX / 4)   // packed backpointer bytes per state
#define NBUF 4            // async gather pipeline depth (prefetch distance 3)

__global__ __launch_bounds__(512)
void FaNat_ctc_align_kernel(const float* __restrict__ ctc,      // [bs,xmax,vocab]
                            const int*   __restrict__ src_size, // [bs]
                            const int*   __restrict__ ys,       // [bs,ymax]
                            const int*   __restrict__ ylens,    // [bs]
                            const int*   __restrict__ blankp,   // [1]
                            float*       __restrict__ out,
                            int xmax, int vocab, int ymax, int bs)
{
    __shared__ float alphaBuf[2][LMAX];           // DP state, double buffered
    __shared__ float lpbuf[NBUF][LMAX];           // async-gather landing zones
    __shared__ int   tokLds[LMAX];                // blank-interleaved path tokens
    __shared__ unsigned char bpp[LMAX * TB4];     // 2-bit backpointers, 4 steps/byte
    __shared__ int   trace[TMAX];                 // states, then csum (reused)

    const int tid   = threadIdx.x;
    const int b     = blockIdx.x;
    const int L     = 2 * ymax + 1;
    const int Traw  = src_size[b];
    const int T     = (Traw < TMAX) ? Traw : TMAX;
    const int blank = blankp[0];
    const int pl    = 2 * ylens[b] + 1;           // path_lens

    // ---- init path tokens and alpha0 ----
    if (tid < L) {
        tokLds[tid] = (tid & 1) ? ys[(size_t)b * ymax + (tid >> 1)] : blank;
        alphaBuf[0][tid] = (tid == 0) ? 0.0f : LOGZERO;
    }
    __syncthreads();

    int  myTok  = 0;
    bool sameTr = false;                          // path[l] == path[l-2]
    if (tid < L) {
        myTok = tokLds[tid];
        if (tid >= 2) sameTr = (myTok == tokLds[tid - 2]);
    }

    // Per-thread gather source: ctc[b, t, tok[l]]
    const float* rowBase = ctc + (size_t)b * xmax * vocab + myTok;
    // LDS byte offsets for async destinations (low 32 bits of generic ptr = LDS offset)
    uint32_t ldsOff[NBUF];
#pragma unroll
    for (int k = 0; k < NBUF; ++k)
        ldsOff[k] = (uint32_t)(uintptr_t)&lpbuf[k][tid];

    // Issue a CDNA5 async global->LDS gather for timestep t into buffer (t % NBUF).
    auto issueGather = [&](int t) {
        if (tid < L) {
            uint64_t ga = (uint64_t)(uintptr_t)(rowBase + (size_t)t * vocab);
            asm volatile("global_load_async_to_lds_b32 %0, %1, off"
                         :: "v"(ldsOff[t & (NBUF - 1)]), "v"(ga)
                         : "memory");
        }
    };

    // Prime the pipeline: gathers for t = 0,1,2
    issueGather(0);
    if (T > 1) issueGather(1);
    if (T > 2) issueGather(2);

    unsigned int bpbits = 0;
    for (int t = 0; t < T; ++t) {
        const int cb = t & 1;                     // alpha parity
        const int gb = t & (NBUF - 1);            // gather buffer for this step
        if (t + 3 < T) {
            // Keep 3 gathers in flight; async loads retire in order, so
            // asynccnt<=3 guarantees step t's data has landed in LDS.
            issueGather(t + 3);
            asm volatile("s_wait_asynccnt 3" ::: "memory");
        } else {
            asm volatile("s_wait_asynccnt 0" ::: "memory");   // tail drain
        }
        __syncthreads();   // all waves drained -> lpbuf[gb] ready, alpha visible

        if (tid < L) {
            const float* ac = alphaBuf[cb];
            float a0 = ac[tid];
            float a1 = (tid >= 1) ? ac[tid - 1] : LOGZERO;
            float a2 = (tid >= 2 && !sameTr) ? ac[tid - 2] : LOGZERO;
            // argmax with first-index tie-break (matches jnp.argmax over [m0,m1,m2])
            int arg; float m;
            if (a0 >= a1 && a0 >= a2) { m = a0; arg = 0; }
            else if (a1 >= a2)        { m = a1; arg = 1; }
            else                      { m = a2; arg = 2; }
            if (tid >= pl) m = LOGZERO;                   // outside path
            alphaBuf[cb ^ 1][tid] = m + lpbuf[gb][tid];
            bpbits |= (unsigned)arg << ((t & 3) * 2);
            if ((t & 3) == 3 || t == T - 1) {
                bpp[tid * TB4 + (t >> 2)] = (unsigned char)bpbits;
                bpbits = 0;
            }
        }
    }
    __syncthreads();   // backpointers + final alpha visible to thread 0

    const size_t YT = (size_t)(ymax + 1) * xmax;

    if (tid == 0) {
        const float* af = alphaBuf[T & 1];        // alpha after src_size steps
        float s1 = af[pl - 1], s2 = af[pl - 2];
        float score = (s1 > s2) ? s1 : s2;
        int   start = (s1 > s2) ? (pl - 1) : (pl - 2);
        out[b] = score;                                       // output 0: score
        out[(size_t)bs + (size_t)bs * YT + b] = (float)(ylens[b] + 1); // output 2

        // ---- backtrace (serial, all in LDS) ----
        int s = start;
        for (int t = T - 1; t >= 0; --t) {
            trace[t] = s;
            if (t > 0) {
                unsigned char by = bpp[s * TB4 + (t >> 2)];
                s -= (int)((by >> ((t & 3) * 2)) & 3);
            }
        }
        // ---- collapse repeats + exclusive cumsum of non-blank tokens ----
        int cnt = 0, prevRaw = 0;                 // shift prepends literal 0
        for (int t = 0; t < T; ++t) {
            int st  = trace[t];
            trace[t] = cnt;                       // csum[t] (exclusive count)
            int raw = tokLds[st];
            int a2v = (raw == prevRaw) ? 0 : raw; // collapse repeats to 0
            prevRaw = raw;
            if (a2v != blank) ++cnt;
        }
    }
    __syncthreads();

    // ---- trigger mask: [ymax+1, xmax] per batch element, as 0.0/1.0 floats ----
    // y-outer / t-inner: fully coalesced row stores, no integer division.
    float* trig = out + bs + (size_t)b * YT;
    for (int y = 0; y <= ymax; ++y) {
        float* row = trig + (size_t)y * xmax;
        for (int t = tid; t < xmax; t += (int)blockDim.x) {
            float v = 0.0f;
            if (t < T) {
                int c = trace[t];
                if (c == y || (y == ymax && t == T - 1)) v = 1.0f;
            }
            row[t] = v;
        }
    }
}

extern "C" void kernel_launch(void* const* d_in, const int* in_sizes, int n_in,
                              void* d_out, int out_size, void* d_ws, size_t ws_size,
                              hipStream_t stream) {
    (void)n_in; (void)d_ws; (void)ws_size; (void)out_size;
    const float* ctc      = (const float*)d_in[0];
    // d_in[1] = src_mask (bool) is derived from src_size; unused.
    const int*   src_size = (const int*)d_in[2];
    const int*   ys       = (const int*)d_in[3];
    const int*   ylens    = (const int*)d_in[4];
    const int*   blank    = (const int*)d_in[5];

    const int bs   = in_sizes[2];
    const int xmax = in_sizes[1] / bs;                       // src_mask = bs*1*xmax
    const int ymax = in_sizes[3] / bs;
    const int vocab = (int)((long long)in_sizes[0] / ((long long)bs * xmax));

    FaNat_ctc_align_kernel<<<bs, 512, 0, stream>>>(
        ctc, src_size, ys, ylens, blank, (float*)d_out, xmax, vocab, ymax, bs);
}